// AdaptiveComputationTime_26001732010492
// MI455X (gfx1250) — compile-verified
//
#include <hip/hip_runtime.h>
#include <hip/hip_bf16.h>
#include <math.h>

#define B_ 8
#define M_ 2048
#define H_ 1024
#define BM (B_ * M_)
#define BMH ((size_t)BM * H_)
#define THRESHOLD_ 0.99f
#define TOK 16
#define ROWSTRIDE 1028  // floats; pad 16 rows of 1024 so strided lane reads hit distinct LDS banks

typedef __attribute__((ext_vector_type(2))) float v2f;
typedef __attribute__((ext_vector_type(8))) float v8f;

// ---------------- K1: rank = cumsum(run, axis=1) - 1 (per batch row) ----------------
__global__ void act_scan_run(const unsigned char* __restrict__ run, int* __restrict__ rank) {
  __shared__ int sums[256];
  const int b = blockIdx.x, tid = threadIdx.x;
  const int base = b * M_ + tid * 8;
  int loc[8];
  int s = 0;
#pragma unroll
  for (int i = 0; i < 8; ++i) { loc[i] = run[base + i] ? 1 : 0; s += loc[i]; }
  sums[tid] = s;
  __syncthreads();
  for (int off = 1; off < 256; off <<= 1) {
    int v = (tid >= off) ? sums[tid - off] : 0;
    __syncthreads();
    if (tid >= off) sums[tid] += v;
    __syncthreads();
  }
  int c = (tid > 0) ? sums[tid - 1] : 0;
#pragma unroll
  for (int i = 0; i < 8; ++i) { c += loc[i]; rank[base + i] = c - 1; }
}

// ---------------- K2: gather h, WMMA gate GEMV, masks, weighted_h_new ----------------
__global__ void act_main(const float* __restrict__ x,
                         const unsigned char* __restrict__ run,
                         const float* __restrict__ acc_p,
                         const float* __restrict__ wh,
                         const float* __restrict__ rem,
                         const int* __restrict__ exit_,
                         const int* __restrict__ updates,
                         const float* __restrict__ p_w,
                         const float* __restrict__ p_b,
                         const int* __restrict__ rank,
                         float* __restrict__ out_wh,
                         float* __restrict__ out_accp,
                         float* __restrict__ out_rem,
                         float* __restrict__ out_runf,
                         float* __restrict__ out_exitf,
                         int* __restrict__ runnew_i) {
  extern __shared__ float smem[];
  float* lds_h   = smem;                      // TOK * ROWSTRIDE
  float* lds_par = smem + TOK * ROWSTRIDE;    // 8 waves * 16 tokens
  float* lds_upd = lds_par + 8 * 16;          // 16
  int*   lds_row = (int*)(lds_upd + 16);      // 16

  const int tid  = threadIdx.x;
  const int tok0 = blockIdx.x * TOK;          // 16 tokens, all in one batch (16 | 2048)
  const int b    = tok0 / M_;

  if (tid < TOK) {
    const int t = tok0 + tid;
    lds_row[tid] = run[t] ? rank[t] : -1;     // unpack source row, -1 => zero row
  }
  __syncthreads();

  // h = unpack(x, run): gather 16 rows into LDS, fully coalesced float4 per row
#pragma unroll 4
  for (int k = 0; k < TOK; ++k) {
    const int row = lds_row[k];
    float4 v = make_float4(0.f, 0.f, 0.f, 0.f);
    if (row >= 0)
      v = ((const float4*)(x + ((size_t)b * M_ + row) * H_))[tid];
    *(float4*)&lds_h[k * ROWSTRIDE + tid * 4] = v;
  }
  __syncthreads();

  // Gate GEMV in full f32 via V_WMMA_F32_16X16X4_F32.
  // A = 16 tokens x 4 K-values; B = 4x16 with every column = p_w slice, so
  // D[m][n] = sum_k h[m][k]*p_w[k] for every n. Each wave owns a 128-wide K slice.
  const int wv = tid >> 5, lane = tid & 31;
  const int laneLo = lane & 15, hiHalf = lane >> 4;
  const float* hrow = &lds_h[laneLo * ROWSTRIDE];
  v8f c = {0.f, 0.f, 0.f, 0.f, 0.f, 0.f, 0.f, 0.f};
#pragma unroll 8
  for (int kk = 0; kk < 32; ++kk) {
    const int k0 = wv * 128 + kk * 4 + 2 * hiHalf;  // A layout: v0 = K0|K2, v1 = K1|K3
    v2f a;  a.x  = hrow[k0];  a.y  = hrow[k0 + 1];
    v2f bm; bm.x = p_w[k0];   bm.y = p_w[k0 + 1];
    c = __builtin_amdgcn_wmma_f32_16x16x4_f32(false, a, false, bm, (short)0, c, false, false);
  }
  // D layout: VGPR r, lanes0-15 -> (M=r, N=lane); lanes16-31 -> (M=r+8). All N equal.
  if (laneLo == 0) {
#pragma unroll
    for (int r = 0; r < 8; ++r) lds_par[wv * 16 + hiHalf * 8 + r] = c[r];
  }
  __syncthreads();

  if (tid < TOK) {
    float s = 0.f;
#pragma unroll
    for (int w2 = 0; w2 < 8; ++w2) s += lds_par[w2 * 16 + tid];
    const int t = tok0 + tid;
    const bool rn = run[t] != 0;
    const float p = rn ? (1.0f / (1.0f + expf(-(s + p_b[0])))) : 0.0f;
    const float ap = acc_p[t];
    const bool mc = rn && ((ap + p) < THRESHOLD_);
    const bool me = rn && !mc;
    const float upd    = mc ? p : (me ? (1.0f - ap) : 0.0f);
    const float ap_new = mc ? (ap + p) : ap;
    lds_upd[tid] = upd;
    out_accp[t]  = ap_new;
    out_rem[t]   = rem[t] + (me ? (1.0f - ap_new) : 0.0f);
    out_runf[t]  = mc ? 1.0f : 0.0f;
    out_exitf[t] = (float)(exit_[t] + (me ? (updates[0] + 1) : 0));
    runnew_i[t]  = mc ? 1 : 0;
  }
  __syncthreads();

  // weighted_h_new = h * update + weighted_h (h==0, upd==0 for non-run tokens)
#pragma unroll 4
  for (int k = 0; k < TOK; ++k) {
    const float u = lds_upd[k];
    const size_t off = ((size_t)(tok0 + k)) * H_ + tid * 4;
    float4 w4 = *(const float4*)(wh + off);
    float4 h4 = *(const float4*)&lds_h[k * ROWSTRIDE + tid * 4];
    float4 o;
    o.x = fmaf(h4.x, u, w4.x);
    o.y = fmaf(h4.y, u, w4.y);
    o.z = fmaf(h4.z, u, w4.z);
    o.w = fmaf(h4.w, u, w4.w);
    *(float4*)(out_wh + off) = o;
  }
}

// ---------------- K3: scan run_new -> inverse permutation + counts ----------------
__global__ void act_scan_runnew(const int* __restrict__ runnew,
                                int* __restrict__ inv, int* __restrict__ counts) {
  __shared__ int sums[256];
  const int b = blockIdx.x, tid = threadIdx.x;
  const int base = tid * 8;
  int loc[8];
  int s = 0;
#pragma unroll
  for (int i = 0; i < 8; ++i) { loc[i] = runnew[b * M_ + base + i]; s += loc[i]; }
  sums[tid] = s;
  __syncthreads();
  for (int off = 1; off < 256; off <<= 1) {
    int v = (tid >= off) ? sums[tid - off] : 0;
    __syncthreads();
    if (tid >= off) sums[tid] += v;
    __syncthreads();
  }
  int c = (tid > 0) ? sums[tid - 1] : 0;
#pragma unroll
  for (int i = 0; i < 8; ++i) {
    if (loc[i]) { inv[b * M_ + c] = base + i; ++c; }
  }
  if (tid == 255) counts[b] = sums[255];
}

// ---------------- K4: h_packed[b,j] = x[b, rank[inv[j]]] if j<count else pad_h ----------------
__global__ void act_pack(const float* __restrict__ x,
                         const float* __restrict__ pad_h,
                         const int* __restrict__ rank,
                         const int* __restrict__ inv,
                         const int* __restrict__ counts,
                         float* __restrict__ out_hp) {
  const int blk = blockIdx.x;       // b*M + j
  const int b = blk / M_;
  const int j = blk - b * M_;
  const int tid = threadIdx.x;      // 256 threads x float4 = 1024 floats
  const float4* src;
  if (j < counts[b]) {
    const int m   = inv[blk];           // original token slot (run_new true => run true)
    const int row = rank[b * M_ + m];   // its unpack source row in x
    src = (const float4*)(x + ((size_t)b * M_ + row) * H_);
  } else {
    src = (const float4*)pad_h;
  }
  const float4 v = src[tid];
  *(float4*)(out_hp + (size_t)blk * H_ + (size_t)tid * 4) = v;
}

extern "C" void kernel_launch(void* const* d_in, const int* in_sizes, int n_in,
                              void* d_out, int out_size, void* d_ws, size_t ws_size,
                              hipStream_t stream) {
  const float*         x       = (const float*)d_in[0];
  const unsigned char* run     = (const unsigned char*)d_in[1];  // jax bool = 1 byte
  const float*         acc_p   = (const float*)d_in[2];
  const float*         wh      = (const float*)d_in[3];
  const float*         rem     = (const float*)d_in[4];
  const int*           exit_   = (const int*)d_in[5];
  const int*           updates = (const int*)d_in[6];
  const float*         pad_h   = (const float*)d_in[7];
  const float*         p_w     = (const float*)d_in[8];
  const float*         p_b     = (const float*)d_in[9];

  int* rank   = (int*)d_ws;     // [B*M]
  int* runnew = rank + BM;      // [B*M]
  int* inv    = runnew + BM;    // [B*M]
  int* counts = inv + BM;       // [B]

  float* out_hp    = (float*)d_out;       // h_packed        [B,M,H]
  float* out_wh    = out_hp + BMH;        // weighted_h_new  [B,M,H]
  float* out_accp  = out_wh + BMH;        // acc_p_new       [B,M,1]
  float* out_rem   = out_accp + BM;       // remainders_new  [B,M,1]
  float* out_runf  = out_rem + BM;        // run_new (0/1)   [B,M]
  float* out_exitf = out_runf + BM;       // exit_new        [B,M,1]

  act_scan_run<<<B_, 256, 0, stream>>>(run, rank);

  const size_t smem = (size_t)(TOK * ROWSTRIDE + 8 * 16 + 16 + 16) * sizeof(float);
  act_main<<<BM / TOK, 256, smem, stream>>>(x, run, acc_p, wh, rem, exit_, updates,
                                            p_w, p_b, rank,
                                            out_wh, out_accp, out_rem, out_runf,
                                            out_exitf, runnew);

  act_scan_runnew<<<B_, 256, 0, stream>>>(runnew, inv, counts);

  act_pack<<<BM, 256, 0, stream>>>(x, pad_h, rank, inv, counts, out_hp);
}